// StnBilinearSampler_74981539053844
// MI455X (gfx1250) — compile-verified
//
#include <hip/hip_runtime.h>
#include <hip/hip_bf16.h>

// STN bilinear sampler for MI455X (gfx1250).
// B=16, H=W=256, C=64, OH=OW=256, all float32.
//
// Memory-bound (~0.5 GFLOP vs ~520 MB HBM -> ~22 us floor @ 23.3 TB/s).
// This version minimizes VALU-per-byte so ALU stays well under the HBM floor:
//   * wave 0 stages the block's 32 (x,y) grid pairs via CDNA5 async DMA
//     (global_load_async_to_lds_b64 + s_wait_asynccnt), then the SAME 32 lanes
//     precompute per-pixel weights (float4) and the 4 clamped corner BYTE
//     offsets (uint4) into LDS -- computed once per pixel, not once per lane.
//   * 256 threads = 32 pixels x 8 lanes; each lane handles channel chunks
//     c and c+8: the 2nd chunk reuses the same corner address with a +128
//     instruction-immediate offset (zero extra address math), giving 8
//     outstanding b128 gathers per thread.
//   * batch index is block-uniform (blocks never straddle an image), so the
//     image base address stays in SGPRs.
//   * output written with non-temporal b128 stores so the streamed 256 MB
//     output does not evict gather-reused images from the 192 MB L2.

typedef float    v4f __attribute__((ext_vector_type(4)));
typedef unsigned v4u __attribute__((ext_vector_type(4)));

#define IMG_H 256
#define IMG_W 256
#define CHN   64
#define PIX_PER_BLOCK 32              // 32 pixels * 8 lanes = 256 threads
#define PIXELS_PER_IMG (IMG_H * IMG_W)   // 65536
#define BLOCKS_PER_IMG (PIXELS_PER_IMG / PIX_PER_BLOCK)  // 2048

__global__ __launch_bounds__(256) void StnBilinearSampler_kernel(
    const float* __restrict__ images,
    const float* __restrict__ grid,
    float* __restrict__ out)
{
    __shared__ __align__(16) float    s_grid[2 * PIX_PER_BLOCK]; // staged (x,y)
    __shared__ __align__(16) float    s_w[PIX_PER_BLOCK][4];     // 4 weights
    __shared__ __align__(16) unsigned s_off[PIX_PER_BLOCK][4];   // 4 corner byte offsets

    const unsigned tid = threadIdx.x;

    // ---- Phase 1 (wave 0 only): async-stage grid pairs, precompute per pixel.
    if (tid < 32u) {
        // Low 32 bits of a generic LDS pointer == AS(3) byte offset.
        unsigned lds_addr = (unsigned)(uintptr_t)(void*)&s_grid[0] + tid * 8u;
        unsigned goff     = (blockIdx.x * 32u + tid) * 8u;   // one (x,y) pair per lane
        asm volatile(
            "global_load_async_to_lds_b64 %0, %1, %2\n\t"
            "s_wait_asynccnt 0"
            :
            : "v"(lds_addr), "v"(goff), "s"(grid)
            : "memory");

        const float gx = s_grid[2u * tid + 0u];
        const float gy = s_grid[2u * tid + 1u];

        // map [-1,1] -> pixel coords
        const float xs = 0.5f * (gx + 1.0f) * (float)(IMG_W - 1);
        const float ys = 0.5f * (gy + 1.0f) * (float)(IMG_H - 1);

        // jnp.round == round-half-to-even == rintf (v_rndne_f32 under RNE)
        const float xr = rintf(xs);
        const float yr = rintf(ys);

        // bilinear weights from UNclamped corner coords (faithful to reference)
        const float fx = xs - xr;
        const float fy = ys - yr;
        s_w[tid][0] = (1.0f - fx) * (1.0f - fy);  // nw
        s_w[tid][1] = fx * (1.0f - fy);           // ne
        s_w[tid][2] = (1.0f - fx) * fy;           // sw
        s_w[tid][3] = fx * fy;                    // se

        // clip after int conversion
        int x0 = (int)xr;     x0 = x0 < 0 ? 0 : (x0 > IMG_W - 1 ? IMG_W - 1 : x0);
        int y0 = (int)yr;     y0 = y0 < 0 ? 0 : (y0 > IMG_H - 1 ? IMG_H - 1 : y0);
        int x1 = (int)xr + 1; x1 = x1 < 0 ? 0 : (x1 > IMG_W - 1 ? IMG_W - 1 : x1);
        int y1 = (int)yr + 1; y1 = y1 < 0 ? 0 : (y1 > IMG_H - 1 ? IMG_H - 1 : y1);

        // byte offsets within this batch image (max 16 MB, fits u32)
        const unsigned rowb = (unsigned)(CHN * 4);        // 256 B per pixel
        s_off[tid][0] = ((unsigned)(y0 * IMG_W + x0)) * rowb;  // nw
        s_off[tid][1] = ((unsigned)(y0 * IMG_W + x1)) * rowb;  // ne
        s_off[tid][2] = ((unsigned)(y1 * IMG_W + x0)) * rowb;  // sw
        s_off[tid][3] = ((unsigned)(y1 * IMG_W + x1)) * rowb;  // se
    }
    __syncthreads();

    // ---- Phase 2: 8 lanes per pixel, each lane does chunks c and c+8.
    const unsigned pix = tid >> 3;     // 0..31
    const unsigned c   = tid & 7u;     // 0..7

    const v4f w = *(const v4f*)s_w[pix];     // broadcast b128 LDS read
    const v4u o = *(const v4u*)s_off[pix];   // corner byte offsets

    // image base for this block's batch: block-uniform -> stays scalar
    const unsigned bimg = blockIdx.x / BLOCKS_PER_IMG;
    const char* base = (const char*)images
                     + (size_t)bimg * ((size_t)IMG_H * IMG_W * CHN * 4)
                     + (size_t)(c * 16u);                  // this lane's chunk

    const char* p_nw = base + o.x;
    const char* p_ne = base + o.y;
    const char* p_sw = base + o.z;
    const char* p_se = base + o.w;

    // 8 coalesced b128 gathers; second chunk via +128B immediate offset
    const v4f nw0 = *(const v4f*)(p_nw);
    const v4f ne0 = *(const v4f*)(p_ne);
    const v4f sw0 = *(const v4f*)(p_sw);
    const v4f se0 = *(const v4f*)(p_se);
    const v4f nw1 = *(const v4f*)(p_nw + 128);
    const v4f ne1 = *(const v4f*)(p_ne + 128);
    const v4f sw1 = *(const v4f*)(p_sw + 128);
    const v4f se1 = *(const v4f*)(p_se + 128);

    v4f acc0 = w.x * nw0;
    acc0 += w.y * ne0;
    acc0 += w.z * sw0;
    acc0 += w.w * se0;

    v4f acc1 = w.x * nw1;
    acc1 += w.y * ne1;
    acc1 += w.z * sw1;
    acc1 += w.w * se1;

    // streamed output: non-temporal b128 stores, keep L2 for image gathers
    const unsigned gpix = blockIdx.x * PIX_PER_BLOCK + pix;
    char* dst = (char*)out + (size_t)gpix * (CHN * 4) + (size_t)(c * 16u);
    __builtin_nontemporal_store(acc0, (v4f*)dst);
    __builtin_nontemporal_store(acc1, (v4f*)(dst + 128));
}

extern "C" void kernel_launch(void* const* d_in, const int* in_sizes, int n_in,
                              void* d_out, int out_size, void* d_ws, size_t ws_size,
                              hipStream_t stream) {
    const float* images = (const float*)d_in[0];   // [16,256,256,64] f32
    const float* grid   = (const float*)d_in[1];   // [16,256,256,2]  f32
    float*       out    = (float*)d_out;           // [16,256,256,64] f32

    const int npix    = in_sizes[1] / 2;           // 16*256*256 = 1,048,576
    const int nblocks = npix / PIX_PER_BLOCK;      // 32,768 blocks of 256 threads

    StnBilinearSampler_kernel<<<dim3(nblocks), dim3(256), 0, stream>>>(images, grid, out);
}